// GATEncoder_34265249088212
// MI455X (gfx1250) — compile-verified
//
#include <hip/hip_runtime.h>
#include <math.h>

#define NEG_SLOPE 0.2f
#define CH 256   // hidden channels (layer-1 out / layer-2 in)
#define CO 128   // output channels

typedef float v2f __attribute__((ext_vector_type(2)));
typedef float v8f __attribute__((ext_vector_type(8)));

// Order-preserving float<->int mapping so we can use integer atomicMax for
// the per-destination softmax max (GLOBAL_ATOMIC_MAX_I32).
__device__ __forceinline__ int f2ord(float f) {
  int i = __float_as_int(f);
  return i >= 0 ? i : (i ^ 0x7FFFFFFF);
}
__device__ __forceinline__ float ord2f(int i) {
  return __int_as_float(i >= 0 ? i : (i ^ 0x7FFFFFFF));
}
__device__ __forceinline__ float lrelu(float v) {
  return v > 0.0f ? v : NEG_SLOPE * v;
}

// ---------------------------------------------------------------- init utils
__global__ void k_init_node(int* __restrict__ maxb, float* __restrict__ denom,
                            float* __restrict__ tnum, int n) {
  int i = blockIdx.x * blockDim.x + threadIdx.x;
  if (i < n) {
    maxb[i]  = (int)0x80000000;  // ordered-int -inf
    denom[i] = 0.0f;
    tnum[i]  = 0.0f;
  }
}

__global__ void k_zero(float* __restrict__ p, int n) {
  int i = blockIdx.x * blockDim.x + threadIdx.x;
  if (i < n) p[i] = 0.0f;
}

// --------------------------------------------------- layer 1 (rank-1 inputs)
// e1 = sum_c lrelu(x[s]*Wl1[c] + x[d]*Wr1[c] + bl1[c]+br1[c]) * att1[c]
__global__ void k_edge1_logits(const float* __restrict__ x,
                               const int* __restrict__ src, const int* __restrict__ dst,
                               int E, int Etot,
                               const float* __restrict__ Wl1, const float* __restrict__ bl1,
                               const float* __restrict__ Wr1, const float* __restrict__ br1,
                               const float* __restrict__ att1,
                               float* __restrict__ e, int* __restrict__ maxb) {
  __shared__ float sWl[CH], sWr[CH], sAtt[CH], sB[CH];
  int t = threadIdx.x;               // blockDim.x == CH == 256
  sWl[t]  = Wl1[t];
  sWr[t]  = Wr1[t];
  sAtt[t] = att1[t];
  sB[t]   = bl1[t] + br1[t];
  __syncthreads();

  int g = blockIdx.x * blockDim.x + t;
  if (g >= Etot) return;
  int s, d;
  if (g < E) { s = src[g]; d = dst[g]; } else { s = d = g - E; }  // self loops
  float xs = x[s], xd = x[d];
  float acc = 0.0f;
#pragma unroll 8
  for (int c = 0; c < CH; ++c) {
    float v = fmaf(xs, sWl[c], fmaf(xd, sWr[c], sB[c]));
    acc = fmaf(lrelu(v), sAtt[c], acc);
  }
  e[g] = acc;
  atomicMax(maxb + d, f2ord(acc));
}

// ex = exp(e - max[d]); denom[d] += ex; tnum[d] += ex * x[s]
__global__ void k_edge1_accum(const float* __restrict__ x,
                              const int* __restrict__ src, const int* __restrict__ dst,
                              int E, int Etot,
                              const float* __restrict__ e, const int* __restrict__ maxb,
                              float* __restrict__ denom, float* __restrict__ tnum) {
  int g = blockIdx.x * blockDim.x + threadIdx.x;
  if (g >= Etot) return;
  int s, d;
  if (g < E) { s = src[g]; d = dst[g]; } else { s = d = g - E; }
  float ex = __expf(e[g] - ord2f(maxb[d]));
  atomicAdd(denom + d, ex);
  atomicAdd(tnum + d, ex * x[s]);
}

// h[i,c] = PReLU( Wl1[c]*(tnum/denom) + bl1[c] + bias1[c] )
__global__ void k_node_h(const float* __restrict__ tnum, const float* __restrict__ denom,
                         const float* __restrict__ Wl1, const float* __restrict__ bl1,
                         const float* __restrict__ bias1, const float* __restrict__ prelu,
                         float* __restrict__ h, int n) {
  int idx = blockIdx.x * blockDim.x + threadIdx.x;
  if (idx >= n * CH) return;
  int i = idx >> 8, c = idx & (CH - 1);
  float t = tnum[i] / denom[i];
  float v = fmaf(Wl1[c], t, bl1[c] + bias1[c]);
  h[idx] = v > 0.0f ? v : prelu[c] * v;
}

// ----------------------------------------------- layer 2 dual GEMM via WMMA
// xl2 = h @ Wl2 + bl2 ; xr2 = h @ Wr2 + br2  (fp32, V_WMMA_F32_16X16X4_F32)
// One wave per 16x16 output tile; A fragment shared between both products.
__global__ void __launch_bounds__(128) k_gemm_dual(
    const float* __restrict__ h, const float* __restrict__ Wl,
    const float* __restrict__ bl, const float* __restrict__ Wr,
    const float* __restrict__ br, float* __restrict__ xl, float* __restrict__ xr,
    int n, int mtiles) {
  int gw = blockIdx.x * (blockDim.x >> 5) + (threadIdx.x >> 5);
  int tm = gw >> 3;        // CO/16 == 8 column tiles
  int tn = gw & 7;
  if (tm >= mtiles) return;

  int lane = threadIdx.x & 31;
  int half = lane >> 4;    // 0: K=0,1  1: K=2,3 per ISA A/B fragment layout
  int l16  = lane & 15;

  int arow = tm * 16 + l16;
  if (arow >= n) arow = n - 1;                 // clamp; stores are guarded
  const float* ap  = h + (size_t)arow * CH + half * 2;
  int col = tn * 16 + l16;
  const float* blp = Wl + col;
  const float* brp = Wr + col;

  v8f accL = {};
  v8f accR = {};
  for (int k = 0; k < CH; k += 4) {
    v2f a = *(const v2f*)(ap + k);             // A[M][k+2h], A[M][k+2h+1]
    int kr = k + half * 2;
    v2f bL, bR;
    bL.x = blp[(size_t)kr * CO];
    bL.y = blp[(size_t)(kr + 1) * CO];
    bR.x = brp[(size_t)kr * CO];
    bR.y = brp[(size_t)(kr + 1) * CO];
    accL = __builtin_amdgcn_wmma_f32_16x16x4_f32(false, a, false, bL, (short)0,
                                                 accL, false, false);
    accR = __builtin_amdgcn_wmma_f32_16x16x4_f32(false, a, false, bR, (short)0,
                                                 accR, false, false);
  }

  float addL = bl[col], addR = br[col];
#pragma unroll
  for (int r = 0; r < 8; ++r) {
    int row = tm * 16 + half * 8 + r;          // C/D layout: M = r + 8*half
    if (row < n) {
      xl[(size_t)row * CO + col] = accL[r] + addL;
      xr[(size_t)row * CO + col] = accR[r] + addR;
    }
  }
}

// ------------------------------------------- layer 2 attention (wave/edge)
__global__ void k_edge2_logits(const float* __restrict__ xl2, const float* __restrict__ xr2,
                               const float* __restrict__ att2,
                               const int* __restrict__ src, const int* __restrict__ dst,
                               int E, int Etot,
                               float* __restrict__ e, int* __restrict__ maxb) {
  int gt = blockIdx.x * blockDim.x + threadIdx.x;
  int gw = gt >> 5;
  int lane = threadIdx.x & 31;
  if (gw >= Etot) return;
  int s, d;
  if (gw < E) { s = src[gw]; d = dst[gw]; } else { s = d = gw - E; }

  float4 a = ((const float4*)(xl2 + (size_t)s * CO))[lane];
  float4 b = ((const float4*)(xr2 + (size_t)d * CO))[lane];
  float4 w = ((const float4*)att2)[lane];
  float acc = lrelu(a.x + b.x) * w.x + lrelu(a.y + b.y) * w.y +
              lrelu(a.z + b.z) * w.z + lrelu(a.w + b.w) * w.w;
#pragma unroll
  for (int off = 16; off > 0; off >>= 1) acc += __shfl_xor(acc, off, 32);
  if (lane == 0) {
    e[gw] = acc;
    atomicMax(maxb + d, f2ord(acc));
  }
}

__global__ void k_edge2_accum(const float* __restrict__ xl2,
                              const int* __restrict__ src, const int* __restrict__ dst,
                              int E, int Etot,
                              const float* __restrict__ e, const int* __restrict__ maxb,
                              float* __restrict__ denom, float* __restrict__ out) {
  int gt = blockIdx.x * blockDim.x + threadIdx.x;
  int gw = gt >> 5;
  int lane = threadIdx.x & 31;
  if (gw >= Etot) return;
  int s, d;
  if (gw < E) { s = src[gw]; d = dst[gw]; } else { s = d = gw - E; }

  float ex = __expf(e[gw] - ord2f(maxb[d]));
  if (lane == 0) atomicAdd(denom + d, ex);
  float4 a = ((const float4*)(xl2 + (size_t)s * CO))[lane];
  float* po = out + (size_t)d * CO + lane * 4;
  atomicAdd(po + 0, ex * a.x);
  atomicAdd(po + 1, ex * a.y);
  atomicAdd(po + 2, ex * a.z);
  atomicAdd(po + 3, ex * a.w);
}

__global__ void k_finalize(float* __restrict__ out, const float* __restrict__ denom,
                           const float* __restrict__ bias2, int n) {
  int idx = blockIdx.x * blockDim.x + threadIdx.x;
  if (idx >= n * CO) return;
  int i = idx >> 7, c = idx & (CO - 1);
  out[idx] = out[idx] / denom[i] + bias2[c];
}

// --------------------------------------------------------------- entry point
extern "C" void kernel_launch(void* const* d_in, const int* in_sizes, int n_in,
                              void* d_out, int out_size, void* d_ws, size_t ws_size,
                              hipStream_t stream) {
  const float* x     = (const float*)d_in[0];
  const int*   ei    = (const int*)d_in[1];
  const float* Wl1   = (const float*)d_in[2];
  const float* bl1   = (const float*)d_in[3];
  const float* Wr1   = (const float*)d_in[4];
  const float* br1   = (const float*)d_in[5];
  const float* att1  = (const float*)d_in[6];
  const float* bias1 = (const float*)d_in[7];
  const float* prelu = (const float*)d_in[8];
  const float* Wl2   = (const float*)d_in[9];
  const float* bl2   = (const float*)d_in[10];
  const float* Wr2   = (const float*)d_in[11];
  const float* br2   = (const float*)d_in[12];
  const float* att2  = (const float*)d_in[13];
  const float* bias2 = (const float*)d_in[14];
  float* out = (float*)d_out;

  int n    = in_sizes[0];        // x is [N,1]
  int E    = in_sizes[1] / 2;    // edge_index is [2,E]
  int Etot = E + n;              // with self loops
  const int* src = ei;
  const int* dst = ei + E;

  // workspace carve-out (fp32): h | xl2 | xr2 | e | maxb | denom | tnum
  float* ws    = (float*)d_ws;
  float* h     = ws;
  float* xl2   = h + (size_t)n * CH;
  float* xr2   = xl2 + (size_t)n * CO;
  float* e     = xr2 + (size_t)n * CO;
  int*   maxb  = (int*)(e + Etot);
  float* denom = (float*)(maxb + n);
  float* tnum  = denom + n;

  int mtiles = (n + 15) / 16;
  dim3 b256(256), b128(128);

  // ---- layer 1 (rank-1 specialization) ----
  k_init_node<<<(n + 255) / 256, b256, 0, stream>>>(maxb, denom, tnum, n);
  k_edge1_logits<<<(Etot + 255) / 256, b256, 0, stream>>>(
      x, src, dst, E, Etot, Wl1, bl1, Wr1, br1, att1, e, maxb);
  k_edge1_accum<<<(Etot + 255) / 256, b256, 0, stream>>>(
      x, src, dst, E, Etot, e, maxb, denom, tnum);
  k_node_h<<<(n * CH + 255) / 256, b256, 0, stream>>>(
      tnum, denom, Wl1, bl1, bias1, prelu, h, n);

  // ---- layer 2 linear transforms (fp32 WMMA) ----
  int waves = mtiles * (CO / 16);
  k_gemm_dual<<<(waves + 3) / 4, b128, 0, stream>>>(
      h, Wl2, bl2, Wr2, br2, xl2, xr2, n, mtiles);

  // ---- layer 2 attention + aggregation ----
  k_init_node<<<(n + 255) / 256, b256, 0, stream>>>(maxb, denom, tnum, n);
  k_zero<<<(n * CO + 255) / 256, b256, 0, stream>>>(out, n * CO);
  int eblocks = (int)(((size_t)Etot * 32 + 255) / 256);
  k_edge2_logits<<<eblocks, b256, 0, stream>>>(
      xl2, xr2, att2, src, dst, E, Etot, e, maxb);
  k_edge2_accum<<<eblocks, b256, 0, stream>>>(
      xl2, src, dst, E, Etot, e, maxb, denom, out);
  k_finalize<<<(n * CO + 255) / 256, b256, 0, stream>>>(out, denom, bias2, n);
}